// TwoDGCNLayer_56753697849745
// MI455X (gfx1250) — compile-verified
//
#include <hip/hip_runtime.h>

// ---------------------------------------------------------------------------
// 2-D Chebyshev GCN layer for MI455X (gfx1250), fp32 WMMA (16x16x4).
//
//   out[b,u,z,o] = relu( sum_{k,l} ( L2[l]^T @ ( (L1[k] @ H[b]) @ W[k,l] ) ) + bias )
//
// Pipeline per batch item b (Hx1 = 48 MiB stays L2-resident between kernels):
//   stage1_gemm : Hx1[k,u,c] = sum_v L1[k,u,v] * H[b,v,c]      (c = w*64+f)
//   stage23     : per u: Q[l] = sum_k Hx1[k,u] @ W[k,l];  out += L2[l]^T @ Q[l]
// ---------------------------------------------------------------------------

typedef __attribute__((ext_vector_type(2))) float v2f;
typedef __attribute__((ext_vector_type(8))) float v8f;

#define NDIM 256              // graph nodes (u,v,w,z)
#define FDIM 64               // features in/out
#define KCH  3                // Chebyshev order
#define CDIM (NDIM * FDIM)    // 16384 flattened (w,f) columns

// D = A(16x4) * B(4x16) + C, fp32.
// 32-bit operand layout (ISA 7.12.2): fragment element j <-> contraction index
// 2*(lane>=16)+j; A row / B col = lane&15; C/D row = r + 8*(lane>=16), col = lane&15.
__device__ __forceinline__ v8f wmma_f32(v2f a, v2f b, v8f c) {
  return __builtin_amdgcn_wmma_f32_16x16x4_f32(
      /*neg_a=*/false, a, /*neg_b=*/false, b,
      /*c_mod=*/(short)0, c, /*reuse_a=*/false, /*reuse_b=*/false);
}

// ---------------------------------------------------------------------------
// Stage 1: Hx1[k,u,c] = sum_v L1[k,u,v] * H[b,v,c]
// grid = (CDIM/512 = 32, NDIM/16 = 16, KCH), block = 256 (8 waves).
// Each wave: one 16(u) x 64(c) output strip, K-loop over v in steps of 4.
// ---------------------------------------------------------------------------
__global__ void __launch_bounds__(256)
stage1_gemm(const float* __restrict__ L1,   // [KCH][NDIM][NDIM]
            const float* __restrict__ Hb,   // [NDIM][CDIM]  (one batch item)
            float* __restrict__ Hx1)        // [KCH][NDIM][CDIM]
{
  const int tid  = threadIdx.x;
  const int lane = tid & 31;
  const int wave = tid >> 5;
  const int m    = lane & 15;   // A row / B col
  const int half = lane >> 4;   // contraction-index half

  const int k  = blockIdx.z;
  const int u0 = blockIdx.y * 16;
  const int c0 = blockIdx.x * 512 + wave * 64;

  const float* Arow = L1 + ((size_t)k * NDIM + (u0 + m)) * NDIM;

  v8f acc0 = {}, acc1 = {}, acc2 = {}, acc3 = {};

  for (int v = 0; v < NDIM; v += 4) {
    const int kr = v + 2 * half;                 // contraction rows kr, kr+1
    v2f a = { Arow[kr], Arow[kr + 1] };          // contiguous float2

    const float* Hr0 = Hb + (size_t)kr * CDIM + c0;
    const float* Hr1 = Hr0 + CDIM;
    v2f b0 = { Hr0[ 0 + m], Hr1[ 0 + m] };
    v2f b1 = { Hr0[16 + m], Hr1[16 + m] };
    v2f b2 = { Hr0[32 + m], Hr1[32 + m] };
    v2f b3 = { Hr0[48 + m], Hr1[48 + m] };

    acc0 = wmma_f32(a, b0, acc0);
    acc1 = wmma_f32(a, b1, acc1);
    acc2 = wmma_f32(a, b2, acc2);
    acc3 = wmma_f32(a, b3, acc3);
  }

  // Store: lane holds column n = m; rows r + 8*half of the 16x16 tile.
#pragma unroll
  for (int r = 0; r < 8; ++r) {
    const int u = u0 + r + 8 * half;
    float* Drow = Hx1 + ((size_t)k * NDIM + u) * CDIM + c0;
    Drow[ 0 + m] = acc0[r];
    Drow[16 + m] = acc1[r];
    Drow[32 + m] = acc2[r];
    Drow[48 + m] = acc3[r];
  }
}

// ---------------------------------------------------------------------------
// Stage 2+3 fused, one block per u (grid = NDIM, block = 256 = 8 waves).
// Streams 16-wide w-tiles:
//   (a) stage Hx1[k, u, w-tile, :] (3*16*64 f32 = 12 KB) into LDS
//   (b) stage A: Q[l][w,o] = sum_k hx[k] @ W[k][l]  -> LDS (12 jobs / 8 waves)
//   (c) stage B: acc[z,o] += L2[l]^T[w-tile] @ Q[l] (persistent 256x64 accum:
//       wave owns z-tiles 2*wave, 2*wave+1  x  4 o-tiles = 64 accum VGPRs/lane)
// Epilogue: bias + ReLU + store.
// ---------------------------------------------------------------------------
__global__ void __launch_bounds__(256)
stage23(const float* __restrict__ Hx1,   // [KCH][NDIM][CDIM]
        const float* __restrict__ L2c,   // [KCH][NDIM][NDIM]  (L2c[l][w][z])
        const float* __restrict__ W,     // [KCH][KCH][FDIM][FDIM]
        const float* __restrict__ bias,  // [FDIM]
        float* __restrict__ outB)        // [NDIM][NDIM][FDIM] (one batch item)
{
  __shared__ __align__(16) float hx[KCH][16][FDIM];  // Hx1 w-tile
  __shared__ __align__(16) float qs[KCH][16][FDIM];  // Q[l] w-tile

  const int tid  = threadIdx.x;
  const int lane = tid & 31;
  const int wave = tid >> 5;
  const int m    = lane & 15;
  const int half = lane >> 4;

  const int u = blockIdx.x;

  v8f acc[2][4] = {};  // [z-subtile][o-tile]

  for (int w0 = 0; w0 < NDIM; w0 += 16) {
    // ---- (a) cooperative Hx1 tile load: 768 float4 across 256 threads ----
#pragma unroll
    for (int p = 0; p < 3; ++p) {
      const int idx = tid + 256 * p;     // float4 index in [0,768)
      const int kk  = idx >> 8;          // cheb k
      const int rem = idx & 255;         // (w, f/4)
      const int w   = rem >> 4;
      const int f   = (rem & 15) * 4;
      const float* src =
          Hx1 + (((size_t)kk * NDIM + u) * NDIM + (w0 + w)) * FDIM + f;
      *((float4*)&hx[0][0][0] + idx) = *(const float4*)src;
    }
    __syncthreads();

    // ---- (b) stage A: Q[l] = sum_k hx[k](16x64) @ W[k][l](64x64) ----
    for (int jj = wave; jj < 12; jj += 8) {      // uniform per wave
      const int l  = jj >> 2;
      const int o0 = (jj & 3) * 16;
      v8f qa = {};
      for (int kc = 0; kc < KCH; ++kc) {
        const float* Wkl = W + ((size_t)kc * KCH + l) * FDIM * FDIM;
#pragma unroll
        for (int s = 0; s < 16; ++s) {           // K = FDIM in steps of 4
          const int f = 4 * s + 2 * half;
          v2f a = { hx[kc][m][f], hx[kc][m][f + 1] };
          v2f b = { Wkl[(size_t)f * FDIM + o0 + m],
                    Wkl[(size_t)(f + 1) * FDIM + o0 + m] };
          qa = wmma_f32(a, b, qa);
        }
      }
#pragma unroll
      for (int r = 0; r < 8; ++r)
        qs[l][r + 8 * half][o0 + m] = qa[r];
    }
    __syncthreads();

    // ---- (c) stage B: acc += L2[l]^T(z,w-tile) @ Q[l](w-tile,o) ----
    for (int l = 0; l < KCH; ++l) {
      const float* L2l = L2c + (size_t)l * NDIM * NDIM;
#pragma unroll
      for (int s = 0; s < 4; ++s) {              // K = 16 in steps of 4
        const int wl = 4 * s + 2 * half;         // local w rows wl, wl+1
        const int wg = w0 + wl;
#pragma unroll
        for (int zt = 0; zt < 2; ++zt) {
          const int z = (2 * wave + zt) * 16 + m;
          v2f a = { L2l[(size_t)wg * NDIM + z],
                    L2l[(size_t)(wg + 1) * NDIM + z] };
#pragma unroll
          for (int ot = 0; ot < 4; ++ot) {
            v2f b = { qs[l][wl][ot * 16 + m], qs[l][wl + 1][ot * 16 + m] };
            acc[zt][ot] = wmma_f32(a, b, acc[zt][ot]);
          }
        }
      }
    }
    // No barrier needed here: next iteration's hx writes are fenced from this
    // iteration's stage-A reads by the post-stage-A barrier, and qs is rewritten
    // only after the next post-load barrier (stage B reads only qs/L2).
  }

  // ---- epilogue: bias + ReLU + store ----
#pragma unroll
  for (int zt = 0; zt < 2; ++zt) {
#pragma unroll
    for (int ot = 0; ot < 4; ++ot) {
      const int o  = ot * 16 + m;
      const float bv = bias[o];
#pragma unroll
      for (int r = 0; r < 8; ++r) {
        const int z = (2 * wave + zt) * 16 + r + 8 * half;
        float val = acc[zt][ot][r] + bv;
        outB[((size_t)u * NDIM + z) * FDIM + o] = val > 0.f ? val : 0.f;
      }
    }
  }
}

// ---------------------------------------------------------------------------
// Launch: per batch item, stage1 then stage23, reusing a 48 MiB Hx1 workspace.
// Stream ordering provides the producer->consumer dependency; no sync calls,
// so this is graph-capture safe and deterministic.
// ---------------------------------------------------------------------------
extern "C" void kernel_launch(void* const* d_in, const int* in_sizes, int n_in,
                              void* d_out, int out_size, void* d_ws, size_t ws_size,
                              hipStream_t stream)
{
  (void)in_sizes; (void)n_in; (void)out_size; (void)ws_size;

  const float* H    = (const float*)d_in[0];  // [8][256][256][64]
  const float* L1   = (const float*)d_in[1];  // [3][256][256]
  const float* L2c  = (const float*)d_in[2];  // [3][256][256]
  const float* W    = (const float*)d_in[3];  // [3][3][64][64]
  const float* bias = (const float*)d_in[4];  // [64]
  float*       out  = (float*)d_out;          // [8][256][256][64]
  float*       Hx1  = (float*)d_ws;           // KCH*NDIM*CDIM f32 = 48 MiB

  for (int b = 0; b < 8; ++b) {
    const float* Hb = H + (size_t)b * NDIM * CDIM;
    dim3 g1(CDIM / 512, NDIM / 16, KCH);
    stage1_gemm<<<g1, 256, 0, stream>>>(L1, Hb, Hx1);

    dim3 g2(NDIM, 1, 1);
    stage23<<<g2, 256, 0, stream>>>(Hx1, L2c, W, bias,
                                    out + (size_t)b * NDIM * NDIM * FDIM);
  }
}